// HebbianLayer_7035156431406
// MI455X (gfx1250) — compile-verified
//
#include <hip/hip_runtime.h>
#include <math.h>

// ---------------------------------------------------------------------------
// HebbianLayer on MI455X (gfx1250): fp32 WMMA GEMM pipeline.
//  - v_wmma_f32_16x16x4_f32 for all matmul FLOPs (~137 GF fp32)
//  - W_new / Wr_new fused into GEMM B-staging (saves 256MB HBM traffic)
//  - double-buffered LDS + register prefetch to overlap HBM with WMMA
//  - k-pair-interleaved B layout: B frag = one aligned ds_load_b64, no shuffles
// ---------------------------------------------------------------------------

typedef __attribute__((ext_vector_type(2))) float v2f;
typedef __attribute__((ext_vector_type(8))) float v8f;

__device__ __forceinline__ float sigmoidf_(float x) { return 1.0f / (1.0f + __expf(-x)); }

// Tile geometry: block = 128 threads = 4 waves; 128x128 block tile, BK=16.
// Each wave owns 64x64 = 4x4 WMMA accumulators.
#define BM 128
#define BN 128
#define BK 16
#define BPITCH (2 * BN + 32)  // 288 floats: 288%64==32 -> halves hit disjoint banks

// ===========================================================================
// Generic WMMA GEMM:  C[M,N] = epi( op(A) @ op(B) )
//   TRA=0: A[m*lda+k]   TRA=1: A[k*lda+m]
//   TRB=0: B[k*ldb+n]   TRB=1: B[n*ldb+k]
//   EPI: 0 none | 1 relu(acc+bias[n]) | 2 acc+bias[n] | 3 acc*scale
// ===========================================================================
template <int TRA, int TRB, int EPI>
__global__ __launch_bounds__(128) void wmma_gemm(
    const float* __restrict__ A, const float* __restrict__ B,
    float* __restrict__ C, const float* __restrict__ bias, float scale,
    int M, int N, int K, int lda, int ldb, int ldc) {
  __shared__ float sA[2][BM][BK + 4];   // +4 pad: column reads hit 16 banks
  __shared__ float sB[2][BK / 2][BPITCH];  // k-pair interleaved: (k,n)->[k/2][2n+(k&1)]

  const int tid = threadIdx.x;
  const int l = tid & 31;
  const int w = tid >> 5;
  const int wr = (w >> 1) * 64, wc = (w & 1) * 64;
  const int lm = l & 15, lh = l >> 4;
  const int m0 = blockIdx.y * BM, n0 = blockIdx.x * BN;

  v8f acc[4][4] = {};
  float4 ra[4], rb[4];

  // ---- register loaders (issue global loads for tile at k0) ----
  auto loadA = [&](int k0) {
    if (TRA == 0) {
#pragma unroll
      for (int it = 0; it < 4; ++it) {
        int s = tid + it * 128;
        int r = s >> 2, c4 = s & 3;
        ra[it] = *(const float4*)(A + (size_t)(m0 + r) * lda + k0 + c4 * 4);
      }
    } else {
#pragma unroll
      for (int it = 0; it < 4; ++it) {
        int s = tid + it * 128;
        int kk = s >> 5, m4 = s & 31;
        ra[it] = *(const float4*)(A + (size_t)(k0 + kk) * lda + m0 + m4 * 4);
      }
    }
  };
  auto loadB = [&](int k0) {
    if (TRB == 0) {
      // slot (kp, ng4): rows k0+2kp, k0+2kp+1 at cols n0+ng4*4 .. +3
#pragma unroll
      for (int it = 0; it < 2; ++it) {
        int s = tid + it * 128;
        int kp = s >> 5, ng4 = s & 31;
        rb[2 * it + 0] = *(const float4*)(B + (size_t)(k0 + 2 * kp) * ldb + n0 + ng4 * 4);
        rb[2 * it + 1] = *(const float4*)(B + (size_t)(k0 + 2 * kp + 1) * ldb + n0 + ng4 * 4);
      }
    } else {
#pragma unroll
      for (int it = 0; it < 4; ++it) {
        int s = tid + it * 128;
        int r = s >> 2, c4 = s & 3;
        rb[it] = *(const float4*)(B + (size_t)(n0 + r) * ldb + k0 + c4 * 4);
      }
    }
  };
  // ---- LDS stores from staged registers ----
  auto storeA = [&](int buf) {
    if (TRA == 0) {
#pragma unroll
      for (int it = 0; it < 4; ++it) {
        int s = tid + it * 128;
        int r = s >> 2, c4 = s & 3;
        sA[buf][r][c4 * 4 + 0] = ra[it].x; sA[buf][r][c4 * 4 + 1] = ra[it].y;
        sA[buf][r][c4 * 4 + 2] = ra[it].z; sA[buf][r][c4 * 4 + 3] = ra[it].w;
      }
    } else {
#pragma unroll
      for (int it = 0; it < 4; ++it) {
        int s = tid + it * 128;
        int kk = s >> 5, m4 = s & 31;
        sA[buf][m4 * 4 + 0][kk] = ra[it].x; sA[buf][m4 * 4 + 1][kk] = ra[it].y;
        sA[buf][m4 * 4 + 2][kk] = ra[it].z; sA[buf][m4 * 4 + 3][kk] = ra[it].w;
      }
    }
  };
  auto storeB = [&](int buf) {
    if (TRB == 0) {
#pragma unroll
      for (int it = 0; it < 2; ++it) {
        int s = tid + it * 128;
        int kp = s >> 5, ng4 = s & 31;
        float4 v0 = rb[2 * it + 0], v1 = rb[2 * it + 1];
        *(float4*)&sB[buf][kp][ng4 * 8 + 0] = make_float4(v0.x, v1.x, v0.y, v1.y);
        *(float4*)&sB[buf][kp][ng4 * 8 + 4] = make_float4(v0.z, v1.z, v0.w, v1.w);
      }
    } else {
#pragma unroll
      for (int it = 0; it < 4; ++it) {
        int s = tid + it * 128;
        int r = s >> 2, c4 = s & 3;  // k = 4*c4 .. 4*c4+3 for column r
        float4 v = rb[it];
        *(float2*)&sB[buf][2 * c4 + 0][2 * r] = make_float2(v.x, v.y);
        *(float2*)&sB[buf][2 * c4 + 1][2 * r] = make_float2(v.z, v.w);
      }
    }
  };

  // ---- software pipeline: prefetch(t+1) overlaps compute(t) ----
  loadA(0);
  loadB(0);
  int buf = 0;
  for (int k0 = 0; k0 < K; k0 += BK) {
    storeA(buf);
    storeB(buf);
    __syncthreads();
    if (k0 + BK < K) { loadA(k0 + BK); loadB(k0 + BK); }  // in flight during WMMAs

#pragma unroll
    for (int kk = 0; kk < BK; kk += 4) {
      v2f a[4], b[4];
#pragma unroll
      for (int mi = 0; mi < 4; ++mi) {
        // A frag: VGPR0 = K {0|2}, VGPR1 = K {1|3} by lane half
        a[mi].x = sA[buf][wr + mi * 16 + lm][kk + 2 * lh];
        a[mi].y = sA[buf][wr + mi * 16 + lm][kk + 2 * lh + 1];
      }
#pragma unroll
      for (int ni = 0; ni < 4; ++ni)
        b[ni] = *(const v2f*)&sB[buf][kk / 2 + lh][2 * (wc + ni * 16 + lm)];
#pragma unroll
      for (int mi = 0; mi < 4; ++mi)
#pragma unroll
        for (int ni = 0; ni < 4; ++ni)
          acc[mi][ni] = __builtin_amdgcn_wmma_f32_16x16x4_f32(
              false, a[mi], false, b[ni], (short)0, acc[mi][ni], false, false);
    }
    __syncthreads();  // all waves done reading buf before it is overwritten
    buf ^= 1;
  }

#pragma unroll
  for (int mi = 0; mi < 4; ++mi)
#pragma unroll
    for (int ni = 0; ni < 4; ++ni)
#pragma unroll
      for (int v = 0; v < 8; ++v) {
        int row = m0 + wr + mi * 16 + v + 8 * lh;
        int col = n0 + wc + ni * 16 + lm;
        float x = acc[mi][ni][v];
        if (EPI == 1) x = fmaxf(x + bias[col], 0.0f);
        else if (EPI == 2) x = x + bias[col];
        else if (EPI == 3) x = x * scale;
        C[(size_t)row * ldc + col] = x;
      }
}

// ===========================================================================
// Fused GEMM vs updated weights:  C = epi( A @ W_new )
//   W_new[k,n] = W[k,n]*(1-sd[k]) + exp(W[k,n]-rowmax[k])*rowsuminv[k]*corr[k,n]
// W/corr tiles prefetched raw into registers; transform applied at LDS-store
// time so its VALU work never stalls the WMMA stream.
//   ADD_RELU=1: C = relu(acc + Cin);  ADD_RELU=0: C = acc
// ===========================================================================
template <int ADD_RELU>
__global__ __launch_bounds__(128) void wmma_gemm_wnew(
    const float* __restrict__ A, const float* __restrict__ W,
    const float* __restrict__ CR, const float* __restrict__ sd,
    const float* __restrict__ rmax, const float* __restrict__ rinv,
    const float* __restrict__ Cin, float* __restrict__ C, int M, int N, int K) {
  __shared__ float sA[2][BM][BK + 4];
  __shared__ float sB[2][BK / 2][BPITCH];

  const int tid = threadIdx.x;
  const int l = tid & 31;
  const int w = tid >> 5;
  const int wr = (w >> 1) * 64, wc = (w & 1) * 64;
  const int lm = l & 15, lh = l >> 4;
  const int m0 = blockIdx.y * BM, n0 = blockIdx.x * BN;

  v8f acc[4][4] = {};
  float4 ra[4], rw[4], rc[4];
  float st_oms[2], st_rm[2], st_ri[2];  // per-slot row stats (k row pair base)

  auto loadT = [&](int k0) {
#pragma unroll
    for (int it = 0; it < 4; ++it) {  // A tile (row-major copy)
      int s = tid + it * 128;
      int r = s >> 2, c4 = s & 3;
      ra[it] = *(const float4*)(A + (size_t)(m0 + r) * K + k0 + c4 * 4);
    }
#pragma unroll
    for (int it = 0; it < 2; ++it) {  // W + corr tiles, raw
      int s = tid + it * 128;
      int kp = s >> 5, ng4 = s & 31;
      int kg = k0 + 2 * kp;
      rw[2 * it + 0] = *(const float4*)(W + (size_t)kg * N + n0 + ng4 * 4);
      rw[2 * it + 1] = *(const float4*)(W + (size_t)(kg + 1) * N + n0 + ng4 * 4);
      rc[2 * it + 0] = *(const float4*)(CR + (size_t)kg * N + n0 + ng4 * 4);
      rc[2 * it + 1] = *(const float4*)(CR + (size_t)(kg + 1) * N + n0 + ng4 * 4);
      st_oms[it] = 1.0f - sd[kg];   // NOTE: per even row; odd row fetched below
      st_rm[it] = rmax[kg];
      st_ri[it] = rinv[kg];
      (void)st_oms; (void)st_rm; (void)st_ri;
    }
  };
  // separate tiny stat fetches for the odd k rows (scalar, L2-resident)
  float st1_oms[2], st1_rm[2], st1_ri[2];
  auto loadStatsOdd = [&](int k0) {
#pragma unroll
    for (int it = 0; it < 2; ++it) {
      int s = tid + it * 128;
      int kp = s >> 5;
      int kg = k0 + 2 * kp + 1;
      st1_oms[it] = 1.0f - sd[kg];
      st1_rm[it] = rmax[kg];
      st1_ri[it] = rinv[kg];
    }
  };

  auto storeT = [&](int buf) {
#pragma unroll
    for (int it = 0; it < 4; ++it) {
      int s = tid + it * 128;
      int r = s >> 2, c4 = s & 3;
      sA[buf][r][c4 * 4 + 0] = ra[it].x; sA[buf][r][c4 * 4 + 1] = ra[it].y;
      sA[buf][r][c4 * 4 + 2] = ra[it].z; sA[buf][r][c4 * 4 + 3] = ra[it].w;
    }
#pragma unroll
    for (int it = 0; it < 2; ++it) {
      int s = tid + it * 128;
      int kp = s >> 5, ng4 = s & 31;
      float4 w0 = rw[2 * it + 0], w1 = rw[2 * it + 1];
      float4 c0 = rc[2 * it + 0], c1 = rc[2 * it + 1];
      float4 o0, o1;
      o0.x = w0.x * st_oms[it] + __expf(w0.x - st_rm[it]) * st_ri[it] * c0.x;
      o0.y = w0.y * st_oms[it] + __expf(w0.y - st_rm[it]) * st_ri[it] * c0.y;
      o0.z = w0.z * st_oms[it] + __expf(w0.z - st_rm[it]) * st_ri[it] * c0.z;
      o0.w = w0.w * st_oms[it] + __expf(w0.w - st_rm[it]) * st_ri[it] * c0.w;
      o1.x = w1.x * st1_oms[it] + __expf(w1.x - st1_rm[it]) * st1_ri[it] * c1.x;
      o1.y = w1.y * st1_oms[it] + __expf(w1.y - st1_rm[it]) * st1_ri[it] * c1.y;
      o1.z = w1.z * st1_oms[it] + __expf(w1.z - st1_rm[it]) * st1_ri[it] * c1.z;
      o1.w = w1.w * st1_oms[it] + __expf(w1.w - st1_rm[it]) * st1_ri[it] * c1.w;
      *(float4*)&sB[buf][kp][ng4 * 8 + 0] = make_float4(o0.x, o1.x, o0.y, o1.y);
      *(float4*)&sB[buf][kp][ng4 * 8 + 4] = make_float4(o0.z, o1.z, o0.w, o1.w);
    }
  };

  loadT(0);
  loadStatsOdd(0);
  int buf = 0;
  for (int k0 = 0; k0 < K; k0 += BK) {
    storeT(buf);
    __syncthreads();
    if (k0 + BK < K) { loadT(k0 + BK); loadStatsOdd(k0 + BK); }

#pragma unroll
    for (int kk = 0; kk < BK; kk += 4) {
      v2f a[4], b[4];
#pragma unroll
      for (int mi = 0; mi < 4; ++mi) {
        a[mi].x = sA[buf][wr + mi * 16 + lm][kk + 2 * lh];
        a[mi].y = sA[buf][wr + mi * 16 + lm][kk + 2 * lh + 1];
      }
#pragma unroll
      for (int ni = 0; ni < 4; ++ni)
        b[ni] = *(const v2f*)&sB[buf][kk / 2 + lh][2 * (wc + ni * 16 + lm)];
#pragma unroll
      for (int mi = 0; mi < 4; ++mi)
#pragma unroll
        for (int ni = 0; ni < 4; ++ni)
          acc[mi][ni] = __builtin_amdgcn_wmma_f32_16x16x4_f32(
              false, a[mi], false, b[ni], (short)0, acc[mi][ni], false, false);
    }
    __syncthreads();
    buf ^= 1;
  }

#pragma unroll
  for (int mi = 0; mi < 4; ++mi)
#pragma unroll
    for (int ni = 0; ni < 4; ++ni)
#pragma unroll
      for (int v = 0; v < 8; ++v) {
        int row = m0 + wr + mi * 16 + v + 8 * lh;
        int col = n0 + wc + ni * 16 + lm;
        float x = acc[mi][ni][v];
        if (ADD_RELU == 1) x = fmaxf(x + Cin[(size_t)row * N + col], 0.0f);
        C[(size_t)row * N + col] = x;
      }
}

// ===========================================================================
// Row softmax over 4096-wide rows; one 256-thread block per row.
// ===========================================================================
__global__ __launch_bounds__(256) void row_softmax(const float* __restrict__ X,
                                                   float* __restrict__ Y) {
  __shared__ float red[256];
  const int row = blockIdx.x, t = threadIdx.x;
  const float* xr = X + (size_t)row * 4096;
  float vals[16];
  float mx = -3.4e38f;
#pragma unroll
  for (int i = 0; i < 16; ++i) { vals[i] = xr[t + i * 256]; mx = fmaxf(mx, vals[i]); }
  red[t] = mx; __syncthreads();
  for (int off = 128; off > 0; off >>= 1) {
    if (t < off) red[t] = fmaxf(red[t], red[t + off]);
    __syncthreads();
  }
  mx = red[0]; __syncthreads();
  float sm = 0.f;
#pragma unroll
  for (int i = 0; i < 16; ++i) { vals[i] = __expf(vals[i] - mx); sm += vals[i]; }
  red[t] = sm; __syncthreads();
  for (int off = 128; off > 0; off >>= 1) {
    if (t < off) red[t] += red[t + off];
    __syncthreads();
  }
  const float inv = 1.0f / red[0];
  float* yr = Y + (size_t)row * 4096;
#pragma unroll
  for (int i = 0; i < 16; ++i) yr[t + i * 256] = vals[i] * inv;
}

// Row softmax stats only: rmax[row], rinv[row] = 1/sum(exp(x-max)).
__global__ __launch_bounds__(256) void row_stats(const float* __restrict__ X,
                                                 float* __restrict__ rmax,
                                                 float* __restrict__ rinv) {
  __shared__ float red[256];
  const int row = blockIdx.x, t = threadIdx.x;
  const float* xr = X + (size_t)row * 4096;
  float vals[16];
  float mx = -3.4e38f;
#pragma unroll
  for (int i = 0; i < 16; ++i) { vals[i] = xr[t + i * 256]; mx = fmaxf(mx, vals[i]); }
  red[t] = mx; __syncthreads();
  for (int off = 128; off > 0; off >>= 1) {
    if (t < off) red[t] = fmaxf(red[t], red[t + off]);
    __syncthreads();
  }
  mx = red[0]; __syncthreads();
  float sm = 0.f;
#pragma unroll
  for (int i = 0; i < 16; ++i) sm += __expf(vals[i] - mx);
  red[t] = sm; __syncthreads();
  for (int off = 128; off > 0; off >>= 1) {
    if (t < off) red[t] += red[t + off];
    __syncthreads();
  }
  if (t == 0) { rmax[row] = mx; rinv[row] = 1.0f / red[0]; }
}

// LayerNorm over 4096-wide rows (population variance, eps=1e-5).
__global__ __launch_bounds__(256) void row_layernorm(
    const float* __restrict__ X, const float* __restrict__ g,
    const float* __restrict__ b, float* __restrict__ Y) {
  __shared__ float rs_[256];
  __shared__ float rq_[256];
  const int row = blockIdx.x, t = threadIdx.x;
  const float* xr = X + (size_t)row * 4096;
  float vals[16];
  float s = 0.f, q = 0.f;
#pragma unroll
  for (int i = 0; i < 16; ++i) {
    vals[i] = xr[t + i * 256];
    s += vals[i];
    q += vals[i] * vals[i];
  }
  rs_[t] = s; rq_[t] = q; __syncthreads();
  for (int off = 128; off > 0; off >>= 1) {
    if (t < off) { rs_[t] += rs_[t + off]; rq_[t] += rq_[t + off]; }
    __syncthreads();
  }
  const float mu = rs_[0] * (1.0f / 4096.0f);
  const float var = rq_[0] * (1.0f / 4096.0f) - mu * mu;
  const float rstd = rsqrtf(var + 1e-5f);
  float* yr = Y + (size_t)row * 4096;
#pragma unroll
  for (int i = 0; i < 16; ++i) {
    int c = t + i * 256;
    yr[c] = (vals[i] - mu) * rstd * g[c] + b[c];
  }
}

// Neuromodulator elementwise: consumes nt [B,3d] interleaved, emits mod_alpha
// and gab (full [B,d] buffer; column-reduced by scaled_decay_k).
__global__ __launch_bounds__(256) void neuromod(
    const float* __restrict__ nt, const float* __restrict__ dopa,
    const float* __restrict__ sero, const float* __restrict__ gaba,
    const float* __restrict__ alpha, float* __restrict__ mod_alpha,
    float* __restrict__ gab_out, int total) {
  int idx = blockIdx.x * blockDim.x + threadIdx.x;
  if (idx >= total) return;
  int j = idx & 4095;
  size_t base = 3 * (size_t)idx;  // nt[i,3j+c] == nt[3*(i*d+j)+c]
  float pd = nt[base], ps = nt[base + 1], pg = nt[base + 2];
  float inv = 1.0f / fmaxf(ps, 1e-6f);
  float dop = tanhf(dopa[idx] + pd * inv);
  float ser = sigmoidf_(sero[idx] + ps);
  float gab = sigmoidf_(gaba[idx] + pg * inv);
  mod_alpha[idx] = alpha[j] * dop * ser;
  gab_out[idx] = gab;
}

// scaled_decay[j] = decay[j] * sigmoid(mean_i gab[i,j])
__global__ __launch_bounds__(256) void scaled_decay_k(
    const float* __restrict__ gab, const float* __restrict__ decay,
    float* __restrict__ sd, int batch) {
  int j = blockIdx.x * blockDim.x + threadIdx.x;
  if (j >= 4096) return;
  float s = 0.f;
  for (int i = 0; i < batch; ++i) s += gab[(size_t)i * 4096 + j];
  sd[j] = decay[j] * sigmoidf_(s / (float)batch);
}

// ===========================================================================
extern "C" void kernel_launch(void* const* d_in, const int* in_sizes, int n_in,
                              void* d_out, int out_size, void* d_ws,
                              size_t ws_size, hipStream_t stream) {
  const float* x        = (const float*)d_in[0];
  const float* prev_act = (const float*)d_in[1];
  const float* prev_rec = (const float*)d_in[2];
  const float* dopamine = (const float*)d_in[3];
  const float* serotonin= (const float*)d_in[4];
  const float* gaba     = (const float*)d_in[5];
  const float* W        = (const float*)d_in[6];
  const float* Wr       = (const float*)d_in[7];
  const float* alpha    = (const float*)d_in[8];
  const float* decay    = (const float*)d_in[9];
  const float* g_act    = (const float*)d_in[10];
  const float* b_act    = (const float*)d_in[11];
  const float* g_rec    = (const float*)d_in[12];
  const float* b_rec    = (const float*)d_in[13];
  const float* p1_w     = (const float*)d_in[14];
  const float* p1_b     = (const float*)d_in[15];
  const float* p2_w     = (const float*)d_in[16];
  const float* p2_b     = (const float*)d_in[17];
  float* out = (float*)d_out;

  float* ws = (float*)d_ws;
  const size_t M1 = 1024 * 1024;
  float* a_soft  = ws;             // [512,4096] softmax(prev_act), later: act
  float* ar_soft = ws + 2 * M1;    // [512,4096] softmax(prev_rec), later: pre_rec
  float* hbuf    = ws + 4 * M1;    // [512,4096] h -> gab -> rec_in (disjoint phases)
  float* nt      = ws + 6 * M1;    // [512,12288]
  float* moda    = ws + 12 * M1;   // [512,4096]
  float* corr    = ws + 14 * M1;   // [4096,4096]
  float* corr_r  = ws + 30 * M1;   // [4096,4096]
  float* small_  = ws + 46 * M1;
  float* wmax = small_;            float* winv  = small_ + 4096;
  float* wrmax = small_ + 8192;    float* wrinv = small_ + 12288;
  float* sd = small_ + 16384;
  float* h = hbuf;
  float* gab = hbuf;
  float* rec_in = hbuf;
  float* pre_rec = ar_soft;
  float* act = a_soft;

  // 1) row softmaxes of prev activations
  row_softmax<<<512, 256, 0, stream>>>(prev_act, a_soft);
  row_softmax<<<512, 256, 0, stream>>>(prev_rec, ar_soft);
  // 2) row softmax stats of W, Wr (feeds fused W_new GEMMs)
  row_stats<<<4096, 256, 0, stream>>>(W, wmax, winv);
  row_stats<<<4096, 256, 0, stream>>>(Wr, wrmax, wrinv);
  // 3) h = relu(prev_act @ p1_w^T + p1_b)   [512,4096,K=4096]
  wmma_gemm<0, 1, 1><<<dim3(32, 4), 128, 0, stream>>>(
      prev_act, p1_w, h, p1_b, 1.0f, 512, 4096, 4096, 4096, 4096, 4096);
  // 4) nt = h @ p2_w^T + p2_b               [512,12288,K=4096]
  wmma_gemm<0, 1, 2><<<dim3(96, 4), 128, 0, stream>>>(
      h, p2_w, nt, p2_b, 1.0f, 512, 12288, 4096, 4096, 4096, 12288);
  // 5) neuromodulators -> mod_alpha, gab
  neuromod<<<8192, 256, 0, stream>>>(nt, dopamine, serotonin, gaba, alpha,
                                     moda, gab, 512 * 4096);
  // 6) scaled decay
  scaled_decay_k<<<16, 256, 0, stream>>>(gab, decay, sd, 512);
  // 7) corr = a_soft^T @ mod_alpha / B      [4096,4096,K=512]
  wmma_gemm<1, 0, 3><<<dim3(32, 32), 128, 0, stream>>>(
      a_soft, moda, corr, nullptr, 1.0f / 512.0f, 4096, 4096, 512, 4096, 4096, 4096);
  wmma_gemm<1, 0, 3><<<dim3(32, 32), 128, 0, stream>>>(
      ar_soft, moda, corr_r, nullptr, 1.0f / 512.0f, 4096, 4096, 512, 4096, 4096, 4096);
  // 8) pre_rec = prev_act @ Wr_new (fused)
  wmma_gemm_wnew<0><<<dim3(32, 4), 128, 0, stream>>>(
      prev_act, Wr, corr_r, sd, wrmax, wrinv, nullptr, pre_rec, 512, 4096, 4096);
  // 9) rec_in = LN(pre_rec; g_rec, b_rec)
  row_layernorm<<<512, 256, 0, stream>>>(pre_rec, g_rec, b_rec, rec_in);
  // 10) act = relu(x @ W_new + rec_in) (fused)
  wmma_gemm_wnew<1><<<dim3(32, 4), 128, 0, stream>>>(
      x, W, corr, sd, wmax, winv, rec_in, act, 512, 4096, 4096);
  // 11) out = LN(act; g_act, b_act)
  row_layernorm<<<512, 256, 0, stream>>>(act, g_act, b_act, out);
}